// loss_uncertainty_71193377898741
// MI455X (gfx1250) — compile-verified
//
#include <hip/hip_runtime.h>
#include <math.h>

typedef __attribute__((ext_vector_type(2))) float v2f;
typedef __attribute__((ext_vector_type(8))) float v8f;

#define NSAMP 16
#define CCH   128
#define HH    64
#define WW    64
#define SHIFT 8
#define NSH   17                    // 2*SHIFT+1
#define NSH2  (NSH*NSH)             // 289
#define SM    (CCH*HH*WW)           // 524288 floats per sample
#define SC    (HH*WW)               // 4096
#define TOTE  (NSAMP*SM)            // 8388608 floats total per tensor

#define YS    16                    // row-space splits per (di, dj-group)

// workspace layout (in floats)
#define OFF_CORR 0
#define N_CORR   (NSH2*NSAMP*NSAMP) // 73984
#define OFF_E    (OFF_CORR + N_CORR)
#define N_E      (NSAMP*SC)         // 65536
#define OFF_PART (OFF_E + N_E)
#define N_PART   (NSAMP*NSH2)       // 4624
#define OFF_S2   (OFF_PART + N_PART)
#define OFF_G2   (OFF_S2 + 16)

__global__ void k_zero(float* __restrict__ p, int n) {
    int i = blockIdx.x * 256 + threadIdx.x;
    if (i < n) p[i] = 0.f;
}

// ---------------------------------------------------------------------------
// corr accumulation. One block: one di (row shift), one dj-group, one slice of
// the (c,h) row space. Per wave: NDJ private 16x16 f32 tiles (v8f each).
// Per (c,h) row each lane loads:
//   G[0..31]  = grd[n, c, h,  32*half .. +31]          (8 aligned b128)
//   S[0..47]  = sat[m, c, hs, 32*half-8 .. +39]        (12 aligned b128, clamped)
// and issues 16 K-chunks x NDJ WMMAs, where the dj shift is a compile-time
// register-pair pick out of S. K-slot mapping per chunk t:
//   K0=2t, K1=2t+1 (lanes 0-15), K2=32+2t, K3=32+2t+1 (lanes 16-31).
// ---------------------------------------------------------------------------
template<int DJLO, int NDJ>
__device__ __forceinline__ void corr_body(const float* __restrict__ sat,
                                          const float* __restrict__ grd,
                                          float* __restrict__ corr,
                                          const int di)
{
    const int lane = threadIdx.x & 31;
    const int wave = threadIdx.x >> 5;
    const int half = lane >> 4;          // 0: lanes 0-15, 1: lanes 16-31
    const int mrow = lane & 15;          // M for A, N for B

    const v8f vzero = {};
    v8f acc[NDJ];
#pragma unroll
    for (int a = 0; a < NDJ; ++a) acc[a] = vzero;

    const int rstart = blockIdx.y * 8 + wave;        // 0 .. YS*8-1
    const int lbase  = mrow * SM + 32 * half;

    for (int r = rstart; r < CCH * HH; r += YS * 8) {
        const int c  = r >> 6;
        const int h  = r & 63;
        const int hs = h + di;
        if (hs < 0 || hs >= HH) continue;            // wave-uniform skip

        float S[48], G[32];

        // sat halo window (boundary handled by clamp + zero-select, no branches)
        const int sbase = lbase + c * SC + hs * WW - 8;
#pragma unroll
        for (int k = 0; k < 12; ++k) {
            int g = sbase + 4 * k;
            g = g < 0 ? 0 : g;
            g = g > (TOTE - 4) ? (TOTE - 4) : g;
            const float4 v = *reinterpret_cast<const float4*>(sat + g);
            S[4*k] = v.x; S[4*k+1] = v.y; S[4*k+2] = v.z; S[4*k+3] = v.w;
        }
#pragma unroll
        for (int i = 0; i < 8; ++i) {                // zero out-of-row halo
            S[i]      = half ? S[i]      : 0.f;      // w < 0   (half 0 only)
            S[40 + i] = half ? 0.f       : S[40+i];  // w >= 64 (half 1 only)
        }

        const int gbase = lbase + c * SC + h * WW;
#pragma unroll
        for (int k = 0; k < 8; ++k) {
            const float4 v = *reinterpret_cast<const float4*>(grd + gbase + 4 * k);
            G[4*k] = v.x; G[4*k+1] = v.y; G[4*k+2] = v.z; G[4*k+3] = v.w;
        }

#pragma unroll
        for (int t = 0; t < 16; ++t) {
            v2f b; b.x = G[2*t]; b.y = G[2*t + 1];
#pragma unroll
            for (int a = 0; a < NDJ; ++a) {
                const int dj = DJLO + a;
                v2f x; x.x = S[8 + 2*t + dj]; x.y = S[9 + 2*t + dj];
                acc[a] = __builtin_amdgcn_wmma_f32_16x16x4_f32(
                             false, x, false, b, (short)0, acc[a], false, false);
            }
        }
    }

    // reduce the 8 wave tiles through LDS, then one global atomic per (s,m,n)
    __shared__ float red[9 * 256];
#pragma unroll
    for (int a = 0; a < NDJ; ++a) red[a * 256 + threadIdx.x] = 0.f;
    __syncthreads();
#pragma unroll
    for (int a = 0; a < NDJ; ++a) {
#pragma unroll
        for (int q = 0; q < 8; ++q) {
            const int m = q + 8 * half;              // C/D layout
            atomicAdd(&red[a * 256 + m * 16 + mrow], acc[a][q]);
        }
    }
    __syncthreads();
    const int di8 = di + SHIFT;
#pragma unroll
    for (int a = 0; a < NDJ; ++a) {
        const int s = di8 * NSH + (DJLO + a + SHIFT);
        atomicAdd(&corr[(size_t)s * 256 + threadIdx.x], red[a * 256 + threadIdx.x]);
    }
}

__global__ __launch_bounds__(256, 1) void k_corr(const float* __restrict__ sat,
                                                 const float* __restrict__ grd,
                                                 float* __restrict__ corr)
{
    const int di = (int)blockIdx.x - SHIFT;
    if (blockIdx.z == 0) corr_body<-8, 9>(sat, grd, corr, di);
    else                 corr_body< 1, 8>(sat, grd, corr, di);
}

// E[m,h,w] = sum_c sat[m,c,h,w]^2
__global__ void k_energy(const float* __restrict__ sat, float* __restrict__ E) {
    int idx = blockIdx.x * blockDim.x + threadIdx.x;
    if (idx >= NSAMP * SC) return;
    int m  = idx >> 12;
    int hw = idx & (SC - 1);
    const float* p = sat + (size_t)m * SM + hw;
    float s = 0.f;
#pragma unroll 4
    for (int c = 0; c < CCH; ++c) { float v = p[(size_t)c * SC]; s += v * v; }
    E[idx] = s;
}

__global__ void k_sumsq_grd(const float* __restrict__ grd, float* __restrict__ G2) {
    int n = blockIdx.x, t = threadIdx.x;
    const float4* p = reinterpret_cast<const float4*>(grd + (size_t)n * SM);
    float s = 0.f;
    for (int i = t; i < SM / 4; i += 256) {
        float4 v = p[i];
        s += v.x * v.x + v.y * v.y + v.z * v.z + v.w * v.w;
    }
    __shared__ float red[256];
    red[t] = s; __syncthreads();
    for (int o = 128; o > 0; o >>= 1) { if (t < o) red[t] += red[t + o]; __syncthreads(); }
    if (!t) G2[n] = red[0];
}

__global__ void k_sum_E(const float* __restrict__ E, float* __restrict__ S2) {
    int m = blockIdx.x, t = threadIdx.x;
    float s = 0.f;
    for (int i = t; i < SC; i += 256) s += E[m * SC + i];
    __shared__ float red[256];
    red[t] = s; __syncthreads();
    for (int o = 128; o > 0; o >>= 1) { if (t < o) red[t] += red[t + o]; __syncthreads(); }
    if (!t) S2[m] = red[0];
}

// partical_raw[m,i,j] = sum of E over the 64x64 window at shift (i,j)
__global__ void k_part(const float* __restrict__ E, float* __restrict__ part) {
    int s = blockIdx.x, m = blockIdx.y, t = threadIdx.x;
    int i = s / NSH, j = s % NSH;
    int h0 = max(0, i - SHIFT), h1 = min(HH, i + HH - SHIFT);
    int w0 = max(0, j - SHIFT), w1 = min(WW, j + WW - SHIFT);
    int nw = w1 - w0, tot = (h1 - h0) * nw;
    float sum = 0.f;
    for (int k = t; k < tot; k += 256) {
        int h = h0 + k / nw, w = w0 + k % nw;
        sum += E[m * SC + h * WW + w];
    }
    __shared__ float red[256];
    red[t] = sum; __syncthreads();
    for (int o = 128; o > 0; o >>= 1) { if (t < o) red[t] += red[t + o]; __syncthreads(); }
    if (!t) part[m * NSH2 + s] = red[0];
}

__device__ inline float softplusf(float x) { return x > 30.f ? x : log1pf(expf(x)); }

// single-block epilogue: normalize, max over shifts, loss + diagnostics
__global__ __launch_bounds__(512) void k_final(const float* __restrict__ corr,
                                               const float* __restrict__ part,
                                               const float* __restrict__ S2,
                                               const float* __restrict__ G2,
                                               float* __restrict__ out)
{
    const int t = threadIdx.x;
    __shared__ float sns[16], sng[16], sdist[256], spos[16], red[512], rmin[16];
    if (t < 16) { sns[t] = sqrtf(S2[t]); sng[t] = sqrtf(G2[t]); }
    __syncthreads();
    if (t < 256) {
        int m = t >> 4, n = t & 15;
        float ns = sns[m], ng = sng[n];
        float inv = 1.f / (ns * ng), ins2 = 1.f / (ns * ns);
        float best = -1e30f;
        for (int s = 0; s < NSH2; ++s) {
            float cr  = corr[(size_t)s * 256 + t] * inv;
            float pn  = part[m * NSH2 + s] * ins2;
            float den = fmaxf(sqrtf(fmaxf(pn, 0.f)), 1e-12f);
            best = fmaxf(best, cr / den);
        }
        sdist[t] = 2.f - 2.f * best;
    }
    __syncthreads();
    if (t < 16) spos[t] = sdist[t * 16 + t];
    __syncthreads();
    float v = 0.f;
    if (t < 256) {
        int m = t >> 4, n = t & 15;
        float d = sdist[t];
        v = softplusf((spos[n] - d) * 10.f)
          + softplusf((spos[m] - d) * 10.f);
    }
    red[t] = v; __syncthreads();
    for (int o = 256; o > 0; o >>= 1) { if (t < o) red[t] += red[t + o]; __syncthreads(); }
    if (t < 16) {
        float mn = 1e30f;
        for (int n = 0; n < 16; ++n) mn = fminf(mn, sdist[t * 16 + n]);
        rmin[t] = mn;
    }
    __syncthreads();
    if (!t) {
        float pair_n = (float)(NSAMP * (NSAMP - 1));
        float loss   = 10.f * 0.5f * red[0] / pair_n;
        float posavg = 0.f, minavg = 0.f;
        for (int k = 0; k < 16; ++k) { posavg += spos[k]; minavg += rmin[k]; }
        out[0] = loss;
        out[1] = posavg / 16.f;
        out[2] = minavg / 16.f;
    }
}

extern "C" void kernel_launch(void* const* d_in, const int* in_sizes, int n_in,
                              void* d_out, int out_size, void* d_ws, size_t ws_size,
                              hipStream_t stream) {
    (void)in_sizes; (void)n_in; (void)out_size; (void)ws_size;
    const float* sat = (const float*)d_in[0];
    const float* grd = (const float*)d_in[1];
    float* ws   = (float*)d_ws;
    float* corr = ws + OFF_CORR;
    float* E    = ws + OFF_E;
    float* part = ws + OFF_PART;
    float* S2   = ws + OFF_S2;
    float* G2   = ws + OFF_G2;

    k_zero  <<<(N_CORR + 255) / 256, 256, 0, stream>>>(corr, N_CORR);
    k_corr  <<<dim3(NSH, YS, 2), 256, 0, stream>>>(sat, grd, corr);
    k_energy<<<(NSAMP * SC) / 256, 256, 0, stream>>>(sat, E);
    k_sumsq_grd<<<16, 256, 0, stream>>>(grd, G2);
    k_sum_E <<<16, 256, 0, stream>>>(E, S2);
    k_part  <<<dim3(NSH2, NSAMP), 256, 0, stream>>>(E, part);
    k_final <<<1, 512, 0, stream>>>(corr, part, S2, G2, (float*)d_out);
}